// Model_64914135712401
// MI455X (gfx1250) — compile-verified
//
#include <hip/hip_runtime.h>

// ODE scan: 10 explicit steps over 16.7M independent elements.
// Memory-bound: 335 MB touch-once traffic -> ~14.4 us at 23.3 TB/s.
// Strategy: fold per-step scalars once (prep kernel), then stream with
// B128 non-temporal loads/stores, packed-f32 (v2f -> v_pk_fma_f32) math,
// and gfx1250 global_prefetch_b8 lookahead.

typedef float v2f __attribute__((ext_vector_type(2)));
typedef float v4f __attribute__((ext_vector_type(4)));

#define NSTEP 10

// Reference constants (computed in double, rounded once to f32)
constexpr float kZB    = 0.01f;
constexpr float kZStep = (float)((0.99 - 0.01) / 10.0);   // linspace spacing
constexpr float kDZ    = (float)((0.99 - 0.01) / 11.0);   // DZ used in update
constexpr float kPT    = (float)((12.0 - 1.0) / 4.0);     // MU = 1 -> 2.75

// ---------------------------------------------------------------------------
// Codegen probe (never launched): does <2 x float> fma select v_pk_fma_f32
// on gfx1250? This function is first in the file so the disasm snippet
// shows its lowering directly.
// ---------------------------------------------------------------------------
__global__ void pk_probe(const v2f* __restrict__ a, const v2f* __restrict__ b,
                         v2f* __restrict__ c) {
  int i = threadIdx.x;
  c[i] = __builtin_elementwise_fma(a[i], b[i], c[i]);
}

// ---------------------------------------------------------------------------
// Main streaming kernel
//   sr' = A1*sr + B + (2*DZ*omr)*si*(sr + inv)
//   si' = A1*si + (DZ*omr)*(S + si^2 - sr'*(sr' + 2*inv)) - G/omr
// cons layout: 10 steps x 8 floats {A1, B, inv, 2*inv, S, G, 0, 0}
// ---------------------------------------------------------------------------
__device__ __forceinline__ v2f splat2(float x) { v2f v = {x, x}; return v; }

__device__ __forceinline__ void ode_pair(v2f& sr, v2f& si, v2f om,
                                         const float* __restrict__ A1,
                                         const float* __restrict__ B,
                                         const float* __restrict__ INV,
                                         const float* __restrict__ TINV,
                                         const float* __restrict__ S,
                                         const float* __restrict__ G) {
  v2f domr  = kDZ * om;
  v2f comr2 = domr + domr;
  v2f invom = splat2(1.0f) / om;      // one precise division per element
#pragma unroll
  for (int s = 0; s < NSTEP; ++s) {
    v2f t   = sr + splat2(INV[s]);
    v2f u   = si * t;
    v2f r0  = __builtin_elementwise_fma(splat2(A1[s]), sr, splat2(B[s]));
    v2f srn = __builtin_elementwise_fma(comr2, u, r0);
    v2f v   = __builtin_elementwise_fma(si, si, splat2(S[s]));
    v2f w   = srn + splat2(TINV[s]);
    v2f P   = __builtin_elementwise_fma(-srn, w, v);
    v2f gq  = splat2(G[s]) * invom;
    v2f q   = __builtin_elementwise_fma(domr, P, -gq);
    si      = __builtin_elementwise_fma(splat2(A1[s]), si, q);
    sr      = srn;
  }
}

__global__ void __launch_bounds__(256)
ode_main(const float* __restrict__ sr_in, const float* __restrict__ si_in,
         const float* __restrict__ omr_in, const float* __restrict__ cons,
         float* __restrict__ sr_out, float* __restrict__ si_out,
         int n, int n4, int stride) {
  const int tid = blockIdx.x * blockDim.x + threadIdx.x;

  // Per-step constants: uniform, invariant loads (SMEM-scalarizable)
  float A1[NSTEP], B[NSTEP], INV[NSTEP], TINV[NSTEP], S[NSTEP], G[NSTEP];
#pragma unroll
  for (int s = 0; s < NSTEP; ++s) {
    float4 c0 = __ldg((const float4*)(cons + s * 8));
    float4 c1 = __ldg((const float4*)(cons + s * 8 + 4));
    A1[s] = c0.x; B[s] = c0.y; INV[s] = c0.z; TINV[s] = c0.w;
    S[s] = c1.x;  G[s] = c1.y;
  }

  const v4f* __restrict__ srp = (const v4f*)sr_in;
  const v4f* __restrict__ sip = (const v4f*)si_in;
  const v4f* __restrict__ omp = (const v4f*)omr_in;
  v4f* __restrict__ srop = (v4f*)sr_out;
  v4f* __restrict__ siop = (v4f*)si_out;

  for (int i4 = tid; i4 < n4; i4 += stride) {
    // gfx1250 global_prefetch_b8 lookahead for next grid-stride chunk
    int nxt = i4 + stride;
    if (nxt < n4) {
      __builtin_prefetch((const void*)(srp + nxt), 0, 0);
      __builtin_prefetch((const void*)(sip + nxt), 0, 0);
      __builtin_prefetch((const void*)(omp + nxt), 0, 0);
    }

    // touch-once streams: non-temporal B128
    v4f sr4 = __builtin_nontemporal_load(srp + i4);
    v4f si4 = __builtin_nontemporal_load(sip + i4);
    v4f om4 = __builtin_nontemporal_load(omp + i4);

    v2f srL = __builtin_shufflevector(sr4, sr4, 0, 1);
    v2f srH = __builtin_shufflevector(sr4, sr4, 2, 3);
    v2f siL = __builtin_shufflevector(si4, si4, 0, 1);
    v2f siH = __builtin_shufflevector(si4, si4, 2, 3);
    v2f omL = __builtin_shufflevector(om4, om4, 0, 1);
    v2f omH = __builtin_shufflevector(om4, om4, 2, 3);

    ode_pair(srL, siL, omL, A1, B, INV, TINV, S, G);
    ode_pair(srH, siH, omH, A1, B, INV, TINV, S, G);

    v4f sro = __builtin_shufflevector(srL, srH, 0, 1, 2, 3);
    v4f sio = __builtin_shufflevector(siL, siH, 0, 1, 2, 3);
    __builtin_nontemporal_store(sro, srop + i4);
    __builtin_nontemporal_store(sio, siop + i4);
  }

  // Scalar tail (n not divisible by 4) — not hit for N = 2^24, kept for safety
  for (int i = (n4 << 2) + tid; i < n; i += stride) {
    float sr = sr_in[i], si = si_in[i], om = omr_in[i];
    float domr = kDZ * om, comr2 = domr + domr, invom = 1.0f / om;
#pragma unroll
    for (int s = 0; s < NSTEP; ++s) {
      float t   = sr + INV[s];
      float srn = fmaf(comr2, si * t, fmaf(A1[s], sr, B[s]));
      float v   = fmaf(si, si, S[s]);
      float P   = fmaf(-srn, srn + TINV[s], v);
      float q   = fmaf(domr, P, -(G[s] * invom));
      si = fmaf(A1[s], si, q);
      sr = srn;
    }
    sr_out[i] = sr;
    si_out[i] = si;
  }
}

// ---------------------------------------------------------------------------
// Prep: fold all (z, fz, fzp)-only terms into 6 per-step scalars.
// ---------------------------------------------------------------------------
__global__ void ode_prep(const float* __restrict__ f_md, float* __restrict__ cons) {
  int i = threadIdx.x;
  if (i >= NSTEP) return;
  float z    = kZB + (float)i * kZStep;
  float fz   = f_md[i];
  float fzp  = (f_md[i + 1] - fz) / kDZ;
  float om   = 1.0f - z;
  float inv  = 1.0f / (kPT * om);
  float inv2 = 1.0f / (kPT * om * om);
  float a    = -fzp / fz;
  float A1   = 1.0f + kDZ * a;
  float B    = kDZ * (a * inv - inv2);
  float tinv = 2.0f * inv;
  float K1   = inv2 / kPT;            // 1/(PT^2 (1-z)^2)
  float K2   = 1.0f / (fz * fz);
  float S    = K2 - K1;
  float G    = kDZ * z * z / fz;      // MU^2 = 1
  float* c = cons + i * 8;
  c[0] = A1; c[1] = B; c[2] = inv; c[3] = tinv; c[4] = S; c[5] = G;
  c[6] = 0.0f; c[7] = 0.0f;
}

// ---------------------------------------------------------------------------
// Launch
// ---------------------------------------------------------------------------
extern "C" void kernel_launch(void* const* d_in, const int* in_sizes, int n_in,
                              void* d_out, int out_size, void* d_ws, size_t ws_size,
                              hipStream_t stream) {
  const float* sr_in = (const float*)d_in[0];   // sig_real (N)
  const float* si_in = (const float*)d_in[1];   // sig_imag (N)
  const float* omr   = (const float*)d_in[2];   // omr      (N)
  const float* f_md  = (const float*)d_in[3];   // f_md     (11)
  const int n = in_sizes[0];

  float* cons   = (float*)d_ws;                 // 10 x 8 floats
  float* sr_out = (float*)d_out;                // tuple (sr, si) concatenated
  float* si_out = sr_out + n;

  ode_prep<<<1, 32, 0, stream>>>(f_md, cons);

  const int n4      = n >> 2;
  const int threads = 256;
  const int iters   = 4;                        // quads per thread
  int blocks = (n4 + threads * iters - 1) / (threads * iters);
  if (blocks < 1) blocks = 1;

  ode_main<<<blocks, threads, 0, stream>>>(sr_in, si_in, omr, cons,
                                           sr_out, si_out, n, n4,
                                           blocks * threads);
}